// GCN_SimpleMultipleOutput_39702677684848
// MI455X (gfx1250) — compile-verified
//
#include <hip/hip_runtime.h>

typedef __attribute__((ext_vector_type(2))) float v2f;
typedef __attribute__((ext_vector_type(8))) float v8f;

#define N_NODES 100000
#define N_EDGES 1600000
#define F_IN    64
#define HIDDEN  64
#define N_OUT   4

// ---------------- utility kernels ----------------

__global__ void k_zero(float* __restrict__ p, int n) {
  int i = blockIdx.x * blockDim.x + threadIdx.x;
  if (i < n) p[i] = 0.0f;
}

__global__ void k_degree(const int* __restrict__ dst, float* __restrict__ deg) {
  int e = blockIdx.x * blockDim.x + threadIdx.x;
  if (e < N_EDGES) unsafeAtomicAdd(&deg[dst[e]], 1.0f);
}

__global__ void k_dinv(const float* __restrict__ deg, float* __restrict__ dinv) {
  int i = blockIdx.x * blockDim.x + threadIdx.x;
  if (i < N_NODES) {
    float d = deg[i];
    dinv[i] = (d > 0.0f) ? rsqrtf(fmaxf(d, 1.0f)) : 0.0f;
  }
}

__global__ void k_norm(const int* __restrict__ src, const int* __restrict__ dst,
                       const float* __restrict__ dinv, float* __restrict__ nrm) {
  int e = blockIdx.x * blockDim.x + threadIdx.x;
  if (e < N_EDGES) nrm[e] = dinv[src[e]] * dinv[dst[e]];
}

__global__ void k_pad_w2(const float* __restrict__ W2, float* __restrict__ W2p) {
  int i = blockIdx.x * blockDim.x + threadIdx.x;  // 64*16 = 1024
  if (i < HIDDEN * 16) {
    int k = i >> 4, c = i & 15;
    W2p[i] = (c < N_OUT) ? W2[k * N_OUT + c] : 0.0f;
  }
}

// ---------------- GEMM 1: XW = X @ W1  (f32 WMMA 16x16x4) ----------------
// Block = 128 threads (4 waves); wave w owns the 16-col tile c0 = 16*w.
// A 16x4 f32 layout: lanes 0-15 hold M=lane, lanes 16-31 hold M=lane-16 with
// K offset +2; vgpr pair = (K, K+1).  B 4x16 mirrored: N = lane&15.
__global__ __launch_bounds__(128) void k_gemm_xw1(const float* __restrict__ X,
                                                  const float* __restrict__ W1,
                                                  float* __restrict__ XW) {
  const int lane = threadIdx.x & 31;
  const int wave = threadIdx.x >> 5;
  const int half = lane >> 4;       // 0 or 1
  const int l16  = lane & 15;
  const int r0   = blockIdx.x * 16; // row tile (100000 % 16 == 0)
  const int c0   = wave * 16;       // col tile
  const int row  = r0 + l16;

  v8f acc = {};
#pragma unroll 4
  for (int k = 0; k < F_IN; k += 4) {
    const int kk = k + 2 * half;
    v2f a = *(const v2f*)(X + row * F_IN + kk);        // A[M=row][kk, kk+1]
    v2f b;
    b.x = W1[kk * HIDDEN + c0 + l16];                  // B[kk][col]
    b.y = W1[(kk + 1) * HIDDEN + c0 + l16];            // B[kk+1][col]
    acc = __builtin_amdgcn_wmma_f32_16x16x4_f32(false, a, false, b,
                                                (short)0, acc, false, false);
  }
#pragma unroll
  for (int r = 0; r < 8; ++r) {
    // C/D: vgpr r -> M = r + 8*half, N = l16
    XW[(r0 + r + 8 * half) * HIDDEN + c0 + l16] = acc[r];
  }
}

// ---------------- Scatter layer 1: agg1[dst] += XW[src] * norm ----------------
// 16 threads per edge, each handles 4 features via a float4 gather + 4 atomics.
__global__ __launch_bounds__(256) void k_scatter1(const int* __restrict__ src,
                                                  const int* __restrict__ dst,
                                                  const float* __restrict__ nrm,
                                                  const float* __restrict__ XW,
                                                  float* __restrict__ agg1) {
  int t = blockIdx.x * blockDim.x + threadIdx.x;
  if (t >= N_EDGES * 16) return;
  int e = t >> 4;
  int f = (t & 15) * 4;
  int s = src[e], d = dst[e];
  float nv = nrm[e];
  float4 v = *(const float4*)(XW + s * HIDDEN + f);
  float* o = agg1 + d * HIDDEN + f;
  unsafeAtomicAdd(o + 0, v.x * nv);
  unsafeAtomicAdd(o + 1, v.y * nv);
  unsafeAtomicAdd(o + 2, v.z * nv);
  unsafeAtomicAdd(o + 3, v.w * nv);
}

// ---------------- GEMM 2: HW2 = relu(agg1 + b1) @ W2pad  (f32 WMMA) ----------
// One wave per 16-row tile; only 16 output cols (4 valid).
__global__ __launch_bounds__(32) void k_gemm_hw2(const float* __restrict__ agg1,
                                                 const float* __restrict__ b1,
                                                 const float* __restrict__ W2p,
                                                 float* __restrict__ HW2) {
  const int lane = threadIdx.x & 31;
  const int half = lane >> 4;
  const int l16  = lane & 15;
  const int r0   = blockIdx.x * 16;
  const int row  = r0 + l16;

  v8f acc = {};
#pragma unroll 4
  for (int k = 0; k < HIDDEN; k += 4) {
    const int kk = k + 2 * half;
    v2f a;
    a.x = fmaxf(agg1[row * HIDDEN + kk]     + b1[kk],     0.0f);   // fused relu+b1
    a.y = fmaxf(agg1[row * HIDDEN + kk + 1] + b1[kk + 1], 0.0f);
    v2f b;
    b.x = W2p[kk * 16 + l16];
    b.y = W2p[(kk + 1) * 16 + l16];
    acc = __builtin_amdgcn_wmma_f32_16x16x4_f32(false, a, false, b,
                                                (short)0, acc, false, false);
  }
  if (l16 < N_OUT) {
#pragma unroll
    for (int r = 0; r < 8; ++r)
      HW2[(r0 + r + 8 * half) * N_OUT + l16] = acc[r];
  }
}

// ---------------- Scatter layer 2: agg2[dst] += HW2[src] * norm --------------
__global__ __launch_bounds__(256) void k_scatter2(const int* __restrict__ src,
                                                  const int* __restrict__ dst,
                                                  const float* __restrict__ nrm,
                                                  const float* __restrict__ HW2,
                                                  float* __restrict__ agg2) {
  int e = blockIdx.x * blockDim.x + threadIdx.x;
  if (e >= N_EDGES) return;
  int s = src[e], d = dst[e];
  float nv = nrm[e];
  float4 v = *(const float4*)(HW2 + s * N_OUT);
  float* o = agg2 + d * N_OUT;
  unsafeAtomicAdd(o + 0, v.x * nv);
  unsafeAtomicAdd(o + 1, v.y * nv);
  unsafeAtomicAdd(o + 2, v.z * nv);
  unsafeAtomicAdd(o + 3, v.w * nv);
}

// ---------------- out = log_softmax(agg2 + b2), duplicated 4x ----------------
__global__ __launch_bounds__(256) void k_lsm(const float* __restrict__ agg2,
                                             const float* __restrict__ b2,
                                             float* __restrict__ out) {
  int n = blockIdx.x * blockDim.x + threadIdx.x;
  if (n >= N_NODES) return;
  float4 v = *(const float4*)(agg2 + n * N_OUT);
  float a0 = v.x + b2[0], a1 = v.y + b2[1], a2 = v.z + b2[2], a3 = v.w + b2[3];
  float m = fmaxf(fmaxf(a0, a1), fmaxf(a2, a3));
  float s = expf(a0 - m) + expf(a1 - m) + expf(a2 - m) + expf(a3 - m);
  float l = m + logf(s);
  float4 r = {a0 - l, a1 - l, a2 - l, a3 - l};
  const int stride = N_NODES * N_OUT;
  *(float4*)(out + 0 * stride + n * 4) = r;
  *(float4*)(out + 1 * stride + n * 4) = r;
  *(float4*)(out + 2 * stride + n * 4) = r;
  *(float4*)(out + 3 * stride + n * 4) = r;
}

// ---------------- host launcher ----------------

extern "C" void kernel_launch(void* const* d_in, const int* in_sizes, int n_in,
                              void* d_out, int out_size, void* d_ws, size_t ws_size,
                              hipStream_t stream) {
  const float* x    = (const float*)d_in[0];
  const int*   ei   = (const int*)d_in[1];     // [2, E]
  const float* W1   = (const float*)d_in[2];
  const float* b1   = (const float*)d_in[3];
  const float* W2   = (const float*)d_in[4];
  const float* b2   = (const float*)d_in[5];
  const int* src = ei;
  const int* dst = ei + N_EDGES;
  float* out = (float*)d_out;

  // workspace layout (floats); zero-needed buffers first & contiguous
  float* ws   = (float*)d_ws;
  float* deg  = ws;                        // N
  float* agg1 = deg  + N_NODES;            // N*64
  float* agg2 = agg1 + N_NODES * HIDDEN;   // N*4
  float* dinv = agg2 + N_NODES * N_OUT;    // N
  float* nrm  = dinv + N_NODES;            // E
  float* XW   = nrm  + N_EDGES;            // N*64
  float* HW2  = XW   + N_NODES * HIDDEN;   // N*4
  float* W2p  = HW2  + N_NODES * N_OUT;    // 64*16

  const int zeroN = N_NODES * (1 + HIDDEN + N_OUT);  // deg+agg1+agg2 contiguous
  k_zero<<<(zeroN + 255) / 256, 256, 0, stream>>>(ws, zeroN);

  k_degree<<<(N_EDGES + 255) / 256, 256, 0, stream>>>(dst, deg);
  k_dinv<<<(N_NODES + 255) / 256, 256, 0, stream>>>(deg, dinv);
  k_norm<<<(N_EDGES + 255) / 256, 256, 0, stream>>>(src, dst, dinv, nrm);
  k_pad_w2<<<4, 256, 0, stream>>>(W2, W2p);

  k_gemm_xw1<<<N_NODES / 16, 128, 0, stream>>>(x, W1, XW);

  k_scatter1<<<(N_EDGES * 16) / 256, 256, 0, stream>>>(src, dst, nrm, XW, agg1);

  k_gemm_hw2<<<N_NODES / 16, 32, 0, stream>>>(agg1, b1, W2p, HW2);

  k_scatter2<<<(N_EDGES + 255) / 256, 256, 0, stream>>>(src, dst, nrm, HW2, agg2);

  k_lsm<<<(N_NODES + 255) / 256, 256, 0, stream>>>(agg2, b2, out);
}